// PhotogrammetricRenderer_11587821765190
// MI455X (gfx1250) — compile-verified
//
#include <hip/hip_runtime.h>
#include <stdint.h>

// NeRF volume render + hierarchical CDF resampling for MI455X (gfx1250).
// Bandwidth-bound (~354 MB @ 23.3 TB/s => ~15us floor). One wave32 per ray;
// TDM (tensor_load_to_lds) stages per-block ray data into LDS; wave scans do
// the cumprod/cumsum; LDS binary search does the importance sampling.

#define NRAYS     65536
#define NSAMP     192
#define NIMP      96
#define RPB       8            // rays (waves) per block
#define CHUNKS    6            // 192 / 32

typedef unsigned int u32x4 __attribute__((ext_vector_type(4)));
typedef int          i32x4 __attribute__((ext_vector_type(4)));
typedef int          i32x8 __attribute__((ext_vector_type(8)));

// Build a D# for a 2D row-major tile (rows x dim0 fp32 elements, stride=dim0)
// and issue TENSOR_LOAD_TO_LDS.  Bit layout per CDNA5 ISA sec 8.3/8.4.
__device__ __forceinline__ void tdm_load_2d(uint32_t lds_byte_off,
                                            const void* gptr,
                                            uint32_t dim0,   // elements per row
                                            uint32_t rows) {
    uint64_t ga = (uint64_t)(uintptr_t)gptr;
    u32x4 g0;
    g0[0] = 1u;                                   // count=1 (valid), user mode
    g0[1] = lds_byte_off;                         // lds_addr
    g0[2] = (uint32_t)(ga & 0xFFFFFFFFu);         // global_addr[31:0]
    g0[3] = (uint32_t)((ga >> 32) & 0x01FFFFFFu)  // global_addr[56:32]
          | (2u << 30);                           // type=2 ("image")
    i32x8 g1;
    g1[0] = (int)(2u << 16);                      // data_size=2 (4B); wg_mask=0
    g1[1] = (int)((dim0 & 0xFFFFu) << 16);        // tensor_dim0[15:0] @ bit48
    g1[2] = (int)(((dim0 >> 16) & 0xFFFFu) |      // tensor_dim0[31:16]
                  ((rows & 0xFFFFu) << 16));      // tensor_dim1[15:0]
    g1[3] = (int)(((rows >> 16) & 0xFFFFu) |      // tensor_dim1[31:16]
                  ((dim0 & 0xFFFFu) << 16));      // tile_dim0
    g1[4] = (int)(rows & 0xFFFFu);                // tile_dim1 (tile_dim2=0)
    g1[5] = (int)dim0;                            // tensor_dim0_stride[31:0]
    g1[6] = 0;                                    // stride hi / dim1_stride lo
    g1[7] = 0;
    i32x4 z4 = {0, 0, 0, 0};
#if defined(__clang_major__) && (__clang_major__ >= 23)
    i32x8 z8 = {0, 0, 0, 0, 0, 0, 0, 0};
    __builtin_amdgcn_tensor_load_to_lds(g0, g1, z4, z4, z8, 0);
#else
    __builtin_amdgcn_tensor_load_to_lds(g0, g1, z4, z4, 0);
#endif
}

__global__ __launch_bounds__(256)
void nerf_render_sample_kernel(const float* __restrict__ densities,
                               const float* __restrict__ colors,
                               const float* __restrict__ z_vals,
                               const float* __restrict__ rays_d,
                               const float* __restrict__ u,
                               float* __restrict__ out) {
    __shared__ float z_lds   [RPB][NSAMP];       //  6 KB
    __shared__ float dn_lds  [RPB][NSAMP];       //  6 KB
    __shared__ float col_lds [RPB][NSAMP * 3];   // 18 KB
    __shared__ float cdf_lds [RPB][NSAMP];       //  6 KB (uses 0..190)
    __shared__ float zmid_lds[RPB][NSAMP];       //  6 KB (uses 0..190)

    const int lane = threadIdx.x & 31;
    const int wv   = threadIdx.x >> 5;           // ray within block
    const int ray0 = blockIdx.x * RPB;
    const int ray  = ray0 + wv;

    // Prefetch this ray's u row while the TDM engine fills LDS.
    __builtin_prefetch(u + (size_t)ray * NIMP, 0, 1);

    if (wv == 0) {
        tdm_load_2d((uint32_t)(uintptr_t)&z_lds[0][0],
                    z_vals    + (size_t)ray0 * NSAMP,     NSAMP,     RPB);
        tdm_load_2d((uint32_t)(uintptr_t)&dn_lds[0][0],
                    densities + (size_t)ray0 * NSAMP,     NSAMP,     RPB);
        tdm_load_2d((uint32_t)(uintptr_t)&col_lds[0][0],
                    colors    + (size_t)ray0 * NSAMP * 3, NSAMP * 3, RPB);
        __builtin_amdgcn_s_wait_tensorcnt(0);
    }
    __syncthreads();

    // ---- per-lane strided samples: s = j*32 + lane ----
    float z[CHUNKS], dn[CHUNKS];
    #pragma unroll
    for (int j = 0; j < CHUNKS; ++j) {
        z[j]  = z_lds[wv][j * 32 + lane];
        dn[j] = dn_lds[wv][j * 32 + lane];
    }

    const float rdx = rays_d[ray * 3 + 0];
    const float rdy = rays_d[ray * 3 + 1];
    const float rdz = rays_d[ray * 3 + 2];
    const float nrm = sqrtf(rdx * rdx + rdy * rdy + rdz * rdz);

    // dists -> alpha, f = 1 - alpha + 1e-10.  z[s+1] via lane shuffles.
    float alpha[CHUNKS], f[CHUNKS], zn1[CHUNKS];
    #pragma unroll
    for (int j = 0; j < CHUNKS; ++j) {
        float zdn = __shfl_down(z[j], 1, 32);                 // z[s+1], lane<31
        int   jn  = (j + 1) % CHUNKS;
        float zch = __shfl(z[jn], 0, 32);                     // next chunk head
        float zp1 = (lane == 31) ? ((j < CHUNKS - 1) ? zch : 0.f) : zdn;
        zn1[j] = zp1;
        float dist = ((j == CHUNKS - 1) && (lane == 31)) ? 1e10f : (zp1 - z[j]);
        dist *= nrm;
        float a = 1.f - __expf(-fmaxf(dn[j], 0.f) * dist);
        alpha[j] = a;
        f[j] = 1.f - a + 1e-10f;
    }

    // ---- transmittance: exclusive cumprod over s via chunked wave scan ----
    float w[CHUNKS];
    float carry = 1.f;
    #pragma unroll
    for (int j = 0; j < CHUNKS; ++j) {
        float incl = f[j];
        #pragma unroll
        for (int off = 1; off < 32; off <<= 1) {
            float t = __shfl_up(incl, off, 32);
            if (lane >= off) incl *= t;
        }
        float excl = __shfl_up(incl, 1, 32);
        float trans = carry * ((lane == 0) ? 1.f : excl);
        w[j] = alpha[j] * trans;
        carry *= __shfl(incl, 31, 32);
    }

    // ---- weighted reductions: rgb, depth, acc ----
    float rr = 0.f, gg = 0.f, bb = 0.f, dep = 0.f, acc = 0.f;
    #pragma unroll
    for (int j = 0; j < CHUNKS; ++j) {
        int s = j * 32 + lane;
        rr  += w[j] * col_lds[wv][3 * s + 0];
        gg  += w[j] * col_lds[wv][3 * s + 1];
        bb  += w[j] * col_lds[wv][3 * s + 2];
        dep += w[j] * z[j];
        acc += w[j];
    }
    #pragma unroll
    for (int off = 16; off >= 1; off >>= 1) {
        rr  += __shfl_xor(rr,  off, 32);
        gg  += __shfl_xor(gg,  off, 32);
        bb  += __shfl_xor(bb,  off, 32);
        dep += __shfl_xor(dep, off, 32);
        acc += __shfl_xor(acc, off, 32);
    }

    // ---- output layout (concatenated tuple, flat) ----
    float* out_rgb   = out;
    float* out_depth = out + (size_t)NRAYS * 3;
    float* out_disp  = out_depth + NRAYS;
    float* out_acc   = out_disp + NRAYS;
    float* out_w     = out_acc + NRAYS;
    float* out_samp  = out_w + (size_t)NRAYS * NSAMP;

    if (lane == 0) {
        out_rgb[ray * 3 + 0] = rr;
        out_rgb[ray * 3 + 1] = gg;
        out_rgb[ray * 3 + 2] = bb;
        out_depth[ray] = dep;
        out_acc[ray]   = acc;
        out_disp[ray]  = 1.f / fmaxf(1e-10f, dep / acc);
    }
    #pragma unroll
    for (int j = 0; j < CHUNKS; ++j)
        out_w[(size_t)ray * NSAMP + j * 32 + lane] = w[j];

    // ---- CDF over w' = weights[1..190] + 1e-5 (190 elems) ----
    float wp[CHUNKS];
    #pragma unroll
    for (int j = 0; j < CHUNKS; ++j) {
        float wdn = __shfl_down(w[j], 1, 32);
        int   jn  = (j + 1) % CHUNKS;
        float wch = __shfl(w[jn], 0, 32);
        float v = (lane == 31) ? ((j < CHUNKS - 1) ? wch : 0.f) : wdn;
        int i = j * 32 + lane;
        wp[j] = (i < NSAMP - 2) ? (v + 1e-5f) : 0.f;
    }
    float wsum = 0.f;
    #pragma unroll
    for (int j = 0; j < CHUNKS; ++j) wsum += wp[j];
    #pragma unroll
    for (int off = 16; off >= 1; off >>= 1) wsum += __shfl_xor(wsum, off, 32);
    const float inv = 1.f / wsum;

    float carryS = 0.f;
    #pragma unroll
    for (int j = 0; j < CHUNKS; ++j) {
        float incl = wp[j];
        #pragma unroll
        for (int off = 1; off < 32; off <<= 1) {
            float t = __shfl_up(incl, off, 32);
            if (lane >= off) incl += t;
        }
        int i = j * 32 + lane;
        if (i < NSAMP - 2) cdf_lds[wv][i + 1] = (carryS + incl) * inv;
        carryS += __shfl(incl, 31, 32);
    }
    if (lane == 0) cdf_lds[wv][0] = 0.f;

    #pragma unroll
    for (int j = 0; j < CHUNKS; ++j) {
        int i = j * 32 + lane;
        if (i < NSAMP - 1) zmid_lds[wv][i] = 0.5f * (z[j] + zn1[j]);
    }
    __syncthreads();

    // ---- importance sampling: searchsorted(cdf, u, right) + lerp ----
    const float* urow = u + (size_t)ray * NIMP;
    float*       srow = out_samp + (size_t)ray * NIMP;
    #pragma unroll
    for (int k = 0; k < NIMP / 32; ++k) {
        float uu = urow[k * 32 + lane];
        int lo = 0, hi = NSAMP - 1;                 // cdf length 191
        while (lo < hi) {
            int mid = (lo + hi) >> 1;
            if (cdf_lds[wv][mid] <= uu) lo = mid + 1; else hi = mid;
        }
        int ind   = lo;                             // in [0, 191]
        int below = ind - 1;
        below = below < 0 ? 0 : (below > NSAMP - 2 ? NSAMP - 2 : below);
        int above = ind > NSAMP - 2 ? NSAMP - 2 : ind;
        float cb = cdf_lds[wv][below], ca = cdf_lds[wv][above];
        float zb = zmid_lds[wv][below], za = zmid_lds[wv][above];
        float den = ca - cb;
        den = (den < 1e-5f) ? 1.f : den;
        float t = (uu - cb) / den;
        srow[k * 32 + lane] = zb + t * (za - zb);
    }
}

extern "C" void kernel_launch(void* const* d_in, const int* in_sizes, int n_in,
                              void* d_out, int out_size, void* d_ws, size_t ws_size,
                              hipStream_t stream) {
    (void)in_sizes; (void)n_in; (void)d_ws; (void)ws_size; (void)out_size;
    const float* densities = (const float*)d_in[0];
    const float* colors    = (const float*)d_in[1];
    const float* z_vals    = (const float*)d_in[2];
    const float* rays_d    = (const float*)d_in[3];
    const float* u         = (const float*)d_in[4];
    float* out = (float*)d_out;

    dim3 grid(NRAYS / RPB);   // 8192 blocks
    dim3 block(32 * RPB);     // 256 threads = 8 wave32 (one ray each)
    hipLaunchKernelGGL(nerf_render_sample_kernel, grid, block, 0, stream,
                       densities, colors, z_vals, rays_d, u, out);
}